// GraphAttentionLayer_61495341744390
// MI455X (gfx1250) — compile-verified
//
#include <hip/hip_runtime.h>
#include <math.h>
#include <stdint.h>

typedef __attribute__((ext_vector_type(2))) float v2f;
typedef __attribute__((ext_vector_type(8))) float v8f;

#define NN   1024
#define NF   128
#define NH   8
#define ND   16
#define SLOPE 0.2f

// Async global->LDS copy, 16 bytes per lane, tracked by ASYNCcnt (CDNA5).
// lds_off = LDS byte offset (low 32 bits of generic shared pointer),
// gaddr   = 64-bit global byte address.
__device__ __forceinline__ void async_b128(uint32_t lds_off, uint64_t gaddr) {
    asm volatile("global_load_async_to_lds_b128 %0, %1, off"
                 :: "v"(lds_off), "v"(gaddr)
                 : "memory");
}

__device__ __forceinline__ void wait_asynccnt0() {
    asm volatile("s_wait_asynccnt 0x0" ::: "memory");
}

// ---------------------------------------------------------------------------
// Kernel 1: g = h @ W^T  (1024x128 = 1024x128 * 128x128^T), fp32 WMMA 16x16x4
// One wave per 16x16 output tile. Block = 256 threads = 8 waves.
// ---------------------------------------------------------------------------
__global__ __launch_bounds__(256) void gemm_hW(const float* __restrict__ hmat,
                                               const float* __restrict__ Wmat,
                                               float* __restrict__ g) {
    const int wave = threadIdx.x >> 5;   // 0..7 -> N tile
    const int lane = threadIdx.x & 31;
    const int m0   = blockIdx.x * 16;
    const int n0   = wave * 16;
    const int r    = lane & 15;
    const int hi   = lane >> 4;          // 0 or 1

    const float* arow = hmat + (m0 + r) * NF + 2 * hi;   // h row, K base 0 or 2
    const float* brow = Wmat + (n0 + r) * NF + 2 * hi;   // W row (B = W^T col)

    v8f c = {};
    #pragma unroll 4
    for (int kk = 0; kk < NF; kk += 4) {
        v2f a = *(const v2f*)(arow + kk);
        v2f b = *(const v2f*)(brow + kk);
        c = __builtin_amdgcn_wmma_f32_16x16x4_f32(
                /*neg_a=*/false, a, /*neg_b=*/false, b,
                /*c_mod=*/(short)0, c, /*reuse_a=*/false, /*reuse_b=*/false);
    }

    float* gout = g + (m0 + 8 * hi) * NF + n0 + r;
    #pragma unroll
    for (int v = 0; v < 8; ++v) gout[v * NF] = c[v];
}

// ---------------------------------------------------------------------------
// Kernel 2: src[i,h] = g[i,h,:].a_src ; dst[i,h] = g[i,h,:].a_dst
// ---------------------------------------------------------------------------
__global__ __launch_bounds__(256) void src_dst(const float* __restrict__ g,
                                               const float* __restrict__ attn_w,
                                               float* __restrict__ srcv,
                                               float* __restrict__ dstv) {
    const int idx = blockIdx.x * 256 + threadIdx.x;   // i*NH + h, 0..8191
    if (idx >= NN * NH) return;
    const float* gv = g + idx * ND;
    float a = 0.f, b = 0.f;
    #pragma unroll
    for (int f = 0; f < ND; ++f) {
        const float gf = gv[f];
        a += gf * attn_w[f];
        b += gf * attn_w[ND + f];
    }
    srcv[idx] = a;
    dstv[idx] = b;
}

// ---------------------------------------------------------------------------
// Kernel 3: fused masked attention, triple softmax, aggregation. One block per
// destination row i. 256 threads: h = tid&7, jslot = tid>>3 (32 lanes/head).
// Phase-0 staging uses CDNA5 async global->LDS B128 copies (ASYNCcnt).
// ---------------------------------------------------------------------------
__global__ __launch_bounds__(256) void gat_row(const float* __restrict__ g,
                                               const float* __restrict__ srcv,
                                               const float* __restrict__ dstv,
                                               const int*   __restrict__ adj,
                                               const float* __restrict__ s,
                                               float* __restrict__ out) {
    __shared__ __align__(16) float eS[NN * NH];  // dst[j,h], then exp(e-m_e) unnorm
    __shared__ __align__(16) float ssS[NN];      // s[i,j], then exp(s-m_s) unnorm
    __shared__ __align__(16) int   adjS[NN];
    __shared__ float accS[NH * ND * 32];         // per-(h,f) partials, 32 jslots
    __shared__ float red[NH * 32];
    __shared__ float redS[32];
    __shared__ float bc_m[NH], bc_inv[NH], bc_mt[NH], bc_invt[NH];
    __shared__ float bc_ms, bc_invs;

    const int i   = blockIdx.x;
    const int tid = threadIdx.x;
    const int hh  = tid & 7;
    const int js  = tid >> 3;            // 0..31

    // ---- phase 0: async-stage row data into LDS (bypasses VGPRs) ----
    {
        const uint32_t eS_off   = (uint32_t)(uintptr_t)&eS[0];
        const uint32_t ssS_off  = (uint32_t)(uintptr_t)&ssS[0];
        const uint32_t adjS_off = (uint32_t)(uintptr_t)&adjS[0];
        const uint64_t dst_ga = (uint64_t)(uintptr_t)dstv;
        const uint64_t s_ga   = (uint64_t)(uintptr_t)(s   + (size_t)i * NN);
        const uint64_t adj_ga = (uint64_t)(uintptr_t)(adj + (size_t)i * NN);

        // dstv: 8192 floats = 2048 x 16B; 256 lanes -> 8 async instructions
        #pragma unroll
        for (int k = 0; k < 8; ++k) {
            const uint32_t b = (uint32_t)(tid + k * 256) * 16u;
            async_b128(eS_off + b, dst_ga + b);
        }
        // s row + adj row: 4KB each = 256 x 16B -> 1 instruction each
        {
            const uint32_t b = (uint32_t)tid * 16u;
            async_b128(ssS_off + b, s_ga + b);
            async_b128(adjS_off + b, adj_ga + b);
        }
        wait_asynccnt0();
    }
    __syncthreads();

    const float srci = srcv[i * NH + hh];

    // ---- pass 1: row maxes of masked leaky_relu(e) and masked s ----
    float me = -INFINITY, ms = -INFINITY;
    for (int k = 0; k < 32; ++k) {
        const int j = js + 32 * k;
        if (adjS[j]) {
            float v = srci + eS[j * NH + hh];
            v = v > 0.f ? v : SLOPE * v;
            me = fmaxf(me, v);
            if (hh == 0) ms = fmaxf(ms, ssS[j]);
        }
    }
    red[hh * 32 + js] = me;
    if (hh == 0) redS[js] = ms;
    __syncthreads();
    if (tid < NH) {
        float m = -INFINITY;
        for (int k = 0; k < 32; ++k) m = fmaxf(m, red[tid * 32 + k]);
        bc_m[tid] = m;
        if (tid == 0) {
            float m2 = -INFINITY;
            for (int k = 0; k < 32; ++k) m2 = fmaxf(m2, redS[k]);
            bc_ms = m2;
        }
    }
    __syncthreads();
    me = bc_m[hh];
    ms = bc_ms;

    // ---- pass 2: sums of exp; cache unnormalized exps in LDS ----
    float se = 0.f, ssum = 0.f;
    for (int k = 0; k < 32; ++k) {
        const int j = js + 32 * k;
        float ex = 0.f;
        if (adjS[j]) {
            float v = srci + eS[j * NH + hh];
            v = v > 0.f ? v : SLOPE * v;
            ex = __expf(v - me);
        }
        eS[j * NH + hh] = ex;               // overwrite dst slot (same thread)
        se += ex;
        if (hh == 0) {
            const float exs = adjS[j] ? __expf(ssS[j] - ms) : 0.f;
            ssS[j] = exs;                   // overwrite s slot (same thread)
            ssum += exs;
        }
    }
    red[hh * 32 + js] = se;
    if (hh == 0) redS[js] = ssum;
    __syncthreads();
    if (tid < NH) {
        float t = 0.f;
        for (int k = 0; k < 32; ++k) t += red[tid * 32 + k];
        bc_inv[tid] = 1.f / t;
        if (tid == 0) {
            float t2 = 0.f;
            for (int k = 0; k < 32; ++k) t2 += redS[k];
            bc_invs = 1.f / t2;
        }
    }
    __syncthreads();
    const float inv_e = bc_inv[hh];
    const float inv_s = bc_invs;

    // ---- pass 3: max over j of t = a + s'  (masked j give exactly 0) ----
    float mt = -INFINITY;
    for (int k = 0; k < 32; ++k) {
        const int j = js + 32 * k;
        const float t = eS[j * NH + hh] * inv_e + ssS[j] * inv_s;
        mt = fmaxf(mt, t);
    }
    red[hh * 32 + js] = mt;
    __syncthreads();
    if (tid < NH) {
        float m = -INFINITY;
        for (int k = 0; k < 32; ++k) m = fmaxf(m, red[tid * 32 + k]);
        bc_mt[tid] = m;
    }
    __syncthreads();
    mt = bc_mt[hh];

    // ---- pass 4: final softmax weights + aggregate g ----
    float st = 0.f;
    float acc[ND];
    #pragma unroll
    for (int f = 0; f < ND; ++f) acc[f] = 0.f;
    for (int k = 0; k < 32; ++k) {
        const int j = js + 32 * k;
        const float w = __expf(eS[j * NH + hh] * inv_e + ssS[j] * inv_s - mt);
        st += w;
        const float4* gp = (const float4*)(g + j * NF + hh * ND);
        #pragma unroll
        for (int q = 0; q < 4; ++q) {
            const float4 gv = gp[q];
            acc[4 * q + 0] += w * gv.x;
            acc[4 * q + 1] += w * gv.y;
            acc[4 * q + 2] += w * gv.z;
            acc[4 * q + 3] += w * gv.w;
        }
    }
    red[hh * 32 + js] = st;
    #pragma unroll
    for (int f = 0; f < ND; ++f) accS[(hh * ND + f) * 32 + js] = acc[f];
    __syncthreads();
    if (tid < NH) {
        float t = 0.f;
        for (int k = 0; k < 32; ++k) t += red[tid * 32 + k];
        bc_invt[tid] = 1.f / t;
    }
    __syncthreads();
    if (tid < NH * ND) {                        // 128 output features of row i
        const int h2 = tid >> 4, f2 = tid & 15;
        float t = 0.f;
        for (int k = 0; k < 32; ++k) t += accS[(h2 * ND + f2) * 32 + k];
        out[i * NF + h2 * ND + f2] = t * bc_invt[h2];
    }
}

// ---------------------------------------------------------------------------
extern "C" void kernel_launch(void* const* d_in, const int* in_sizes, int n_in,
                              void* d_out, int out_size, void* d_ws, size_t ws_size,
                              hipStream_t stream) {
    const float* h_in   = (const float*)d_in[0];   // (1024,128)
    const int*   adj    = (const int*)  d_in[1];   // (1024,1024,1)
    const float* s_in   = (const float*)d_in[2];   // (1024,1024,1)
    const float* W_in   = (const float*)d_in[3];   // (128,128)
    const float* attn_w = (const float*)d_in[4];   // (32,)
    float* out = (float*)d_out;                    // (1024,128)

    float* g    = (float*)d_ws;                    // 1024*128 floats
    float* srcv = g + NN * NF;                     // 1024*8
    float* dstv = srcv + NN * NH;                  // 1024*8

    gemm_hW<<<NN / 16, 256, 0, stream>>>(h_in, W_in, g);
    src_dst<<<(NN * NH) / 256, 256, 0, stream>>>(g, attn_w, srcv, dstv);
    gat_row<<<NN, 256, 0, stream>>>(g, srcv, dstv, adj, s_in, out);
}